// LearnableLowpass_67740224193134
// MI455X (gfx1250) — compile-verified
//
#include <hip/hip_runtime.h>
#include <stdint.h>

// ---------------- problem constants ----------------
#define SR_F    48000.0f
#define BATCH   32
#define TLEN    480000
#define CHUNKS  960           // chunks per batch row
#define LCH     500           // TLEN / CHUNKS
#define TILE    100           // elements per TDM tile column-chunk (400B, mult of 4)
#define NTILES  5             // LCH / TILE
#define ROWS    32            // chunks handled per wave == lanes
#define WPR     (CHUNKS/ROWS) // waves per batch row = 30
#define SEG     (CHUNKS/32)   // chunks per lane in the prefix-scan kernel = 30
#define TILE_FLOATS (ROWS*TILE)     // 3200 floats
#define TILE_BYTES  (TILE_FLOATS*4) // 12800 bytes
#define PI_F 3.14159265358979323846f

typedef unsigned int v4u __attribute__((ext_vector_type(4)));
typedef int          v4i __attribute__((ext_vector_type(4)));
typedef int          v8i __attribute__((ext_vector_type(8)));

#if defined(__has_builtin)
#if __has_builtin(__builtin_amdgcn_tensor_load_to_lds) && \
    __has_builtin(__builtin_amdgcn_tensor_store_from_lds) && \
    __has_builtin(__builtin_amdgcn_s_wait_tensorcnt)
#define HAVE_TDM 1
#else
#define HAVE_TDM 0
#endif
#else
#define HAVE_TDM 0
#endif

// ---------------- biquad coefficients (torchaudio lowpass_biquad) ----------------
__device__ __forceinline__ void lp_coeffs(float f_in, float q_in,
                                          float& b0, float& b1, float& b2,
                                          float& a1, float& a2) {
  float f = fminf(fmaxf(f_in, 100.0f), SR_F * 0.5f - 1.0f);
  float q = fminf(fmaxf(q_in, 0.1f), 10.0f);
  float w0 = 2.0f * PI_F * f / SR_F;
  float alpha = __sinf(w0) / (2.0f * q);
  float cw = __cosf(w0);
  float a0 = 1.0f + alpha;
  b0 = ((1.0f - cw) * 0.5f) / a0;
  b1 = (1.0f - cw) / a0;
  b2 = b0;
  a1 = (-2.0f * cw) / a0;
  a2 = (1.0f - alpha) / a0;
}

// A = [[-a1,-a2],[1,0]] ; compute A^n in double via binary exponentiation
__device__ __forceinline__ void mat_pow(double a1, double a2, int n,
                                        double& m00, double& m01,
                                        double& m10, double& m11) {
  double r00 = 1, r01 = 0, r10 = 0, r11 = 1;
  double p00 = -a1, p01 = -a2, p10 = 1, p11 = 0;
  while (n > 0) {
    if (n & 1) {
      double t00 = r00 * p00 + r01 * p10, t01 = r00 * p01 + r01 * p11;
      double t10 = r10 * p00 + r11 * p10, t11 = r10 * p01 + r11 * p11;
      r00 = t00; r01 = t01; r10 = t10; r11 = t11;
    }
    double s00 = p00 * p00 + p01 * p10, s01 = p00 * p01 + p01 * p11;
    double s10 = p10 * p00 + p11 * p10, s11 = p10 * p01 + p11 * p11;
    p00 = s00; p01 = s01; p10 = s10; p11 = s11;
    n >>= 1;
  }
  m00 = r00; m01 = r01; m10 = r10; m11 = r11;
}

// ---------------- TDM descriptor (CDNA5 ISA) ----------------
#if HAVE_TDM
struct TdmDesc { v4u g0; v8i g1; };

__device__ __forceinline__ TdmDesc make_desc(uint64_t gaddr, uint32_t lds_off) {
  TdmDesc d;
  // Group 0: count=1 | lds_addr | global_addr[56:0] | type=2
  d.g0[0] = 1u;
  d.g0[1] = lds_off;
  d.g0[2] = (uint32_t)(gaddr & 0xffffffffu);
  d.g0[3] = (uint32_t)((gaddr >> 32) & 0x01ffffffu) | (2u << 30);
  // Group 1: 2D tensor, data_size=4B, tile = TILE x ROWS, row stride LCH elems
  const uint32_t dim0 = TILE, dim1 = ROWS, stride0 = LCH;
  uint32_t w0 = (2u << 16);                                       // data_size=4B, wg_mask=0
  uint32_t w1 = (dim0 & 0xffffu) << 16;                           // tensor_dim0[15:0]
  uint32_t w2 = ((dim0 >> 16) & 0xffffu) | ((dim1 & 0xffffu) << 16);
  uint32_t w3 = ((dim1 >> 16) & 0xffffu) | ((uint32_t)TILE << 16); // tile_dim0
  uint32_t w4 = (uint32_t)ROWS;                                    // tile_dim1=32, tile_dim2=0
  uint32_t w5 = stride0;                                           // tensor_dim0_stride[31:0]
  uint32_t w6 = 0, w7 = 0;
  d.g1[0] = (int)w0; d.g1[1] = (int)w1; d.g1[2] = (int)w2; d.g1[3] = (int)w3;
  d.g1[4] = (int)w4; d.g1[5] = (int)w5; d.g1[6] = (int)w6; d.g1[7] = (int)w7;
  return d;
}

__device__ __forceinline__ void tdm_load(uint64_t gaddr, uint32_t lds_off) {
  TdmDesc d = make_desc(gaddr, lds_off);
  v4i z4 = {0, 0, 0, 0};
#if __clang_major__ >= 23
  v8i z8 = {0, 0, 0, 0, 0, 0, 0, 0};
  __builtin_amdgcn_tensor_load_to_lds(d.g0, d.g1, z4, z4, z8, 0);
#else
  __builtin_amdgcn_tensor_load_to_lds(d.g0, d.g1, z4, z4, 0);
#endif
}

__device__ __forceinline__ void tdm_store(uint64_t gaddr, uint32_t lds_off) {
  TdmDesc d = make_desc(gaddr, lds_off);
  v4i z4 = {0, 0, 0, 0};
#if __clang_major__ >= 23
  v8i z8 = {0, 0, 0, 0, 0, 0, 0, 0};
  __builtin_amdgcn_tensor_store_from_lds(d.g0, d.g1, z4, z4, z8, 0);
#else
  __builtin_amdgcn_tensor_store_from_lds(d.g0, d.g1, z4, z4, 0);
#endif
}
#endif // HAVE_TDM

// one biquad step, depth-2 FMA critical path
#define BIQ_STEP(xt)                                            \
  {                                                             \
    float xp = fmaf(b0, (xt), fmaf(b1, x1, b2 * x2));           \
    float yv = fmaf(na1, y1, fmaf(na2, y2, xp));                \
    x2 = x1; x1 = (xt); y2 = y1; y1 = yv;                       \
  }

// ============ Kernel 1: per-chunk zero-y-state offsets d = (y_end, y_end-1) ============
__global__ void k1_chunk_state(const float* __restrict__ x,
                               const float* __restrict__ ff,
                               const float* __restrict__ fq,
                               float* __restrict__ d_state) {
  extern __shared__ float smem[];
  const int lane = threadIdx.x;
  const int b = blockIdx.x / WPR;
  const int w = blockIdx.x % WPR;
  const int c = w * ROWS + lane;

  float b0, b1, b2, a1, a2;
  lp_coeffs(ff[0], fq[0], b0, b1, b2, a1, a2);
  const float na1 = -a1, na2 = -a2;

  const uint64_t rowBase = (uint64_t)b * TLEN;
  const uint64_t chunkStart = rowBase + (uint64_t)c * LCH;
  float x1 = 0.f, x2 = 0.f;
  if (c > 0) { x1 = x[chunkStart - 1]; x2 = x[chunkStart - 2]; }
  float y1 = 0.f, y2 = 0.f;

#if HAVE_TDM
  const uint64_t xbase = (uint64_t)(uintptr_t)x + 4ull * (rowBase + (uint64_t)(w * ROWS) * LCH);
  tdm_load(xbase, 0u);
  for (int t = 0; t < NTILES; ++t) {
    if (t + 1 < NTILES) {
      asm volatile("s_wait_dscnt 0" ::: "memory");
      tdm_load(xbase + 4ull * (uint64_t)(t + 1) * TILE,
               ((t + 1) & 1) ? (uint32_t)TILE_BYTES : 0u);
      __builtin_amdgcn_s_wait_tensorcnt((short)1);
    } else {
      __builtin_amdgcn_s_wait_tensorcnt((short)0);
    }
    asm volatile("" ::: "memory");
    const float4* bv = (const float4*)(smem + (t & 1) * TILE_FLOATS + lane * TILE);
    for (int j = 0; j < TILE / 4; ++j) {
      float4 v = bv[j];
      BIQ_STEP(v.x); BIQ_STEP(v.y); BIQ_STEP(v.z); BIQ_STEP(v.w);
    }
  }
#else
  const float* gx = x + chunkStart;
  for (int n = 0; n < LCH; ++n) { float xt = gx[n]; BIQ_STEP(xt); }
#endif

  const int idx = (b * CHUNKS + c) * 2;
  d_state[idx] = y1;
  d_state[idx + 1] = y2;
}

// ============ Kernel 2: affine prefix scan of chunk transforms (per batch row) ============
__global__ void k2_prefix(const float* __restrict__ ff,
                          const float* __restrict__ fq,
                          const float* __restrict__ d_state,
                          float* __restrict__ s_init) {
  const int b = blockIdx.x;
  const int lane = threadIdx.x;

  float b0, b1, b2, a1, a2;
  lp_coeffs(ff[0], fq[0], b0, b1, b2, a1, a2);

  double m00d, m01d, m10d, m11d;
  mat_pow((double)a1, (double)a2, LCH, m00d, m01d, m10d, m11d);          // per-chunk M = A^L
  const float M00 = (float)m00d, M01 = (float)m01d, M10 = (float)m10d, M11 = (float)m11d;
  mat_pow((double)a1, (double)a2, LCH * SEG, m00d, m01d, m10d, m11d);    // per-segment A^(L*SEG)
  float P00 = (float)m00d, P01 = (float)m01d, P10 = (float)m10d, P11 = (float)m11d;

  const float* dbase = d_state + (size_t)(b * CHUNKS) * 2;
  const int c0 = lane * SEG;

  // phase A: lane's segment with zero incoming state -> v
  float v1 = 0.f, v2 = 0.f;
  for (int k = 0; k < SEG; ++k) {
    float d1 = dbase[(c0 + k) * 2], d2 = dbase[(c0 + k) * 2 + 1];
    float n1 = fmaf(M00, v1, fmaf(M01, v2, d1));
    float n2 = fmaf(M10, v1, fmaf(M11, v2, d2));
    v1 = n1; v2 = n2;
  }

  // phase B: inclusive Hillis-Steele scan of affine pairs (P, v) across 32 lanes
  for (int off = 1; off < 32; off <<= 1) {
    float q00 = __shfl_up(P00, off), q01 = __shfl_up(P01, off);
    float q10 = __shfl_up(P10, off), q11 = __shfl_up(P11, off);
    float w1 = __shfl_up(v1, off),   w2 = __shfl_up(v2, off);
    float nv1 = fmaf(P00, w1, fmaf(P01, w2, v1));
    float nv2 = fmaf(P10, w1, fmaf(P11, w2, v2));
    float n00 = P00 * q00 + P01 * q10, n01 = P00 * q01 + P01 * q11;
    float n10 = P10 * q00 + P11 * q10, n11 = P10 * q01 + P11 * q11;
    if (lane >= off) {
      P00 = n00; P01 = n01; P10 = n10; P11 = n11;
      v1 = nv1; v2 = nv2;
    }
  }

  // exclusive: incoming state of this lane's segment (global init state = 0 -> just v)
  float s1 = __shfl_up(v1, 1), s2 = __shfl_up(v2, 1);
  if (lane == 0) { s1 = 0.f; s2 = 0.f; }

  // phase C: replay segment storing per-chunk initial states
  for (int k = 0; k < SEG; ++k) {
    const int idx = (b * CHUNKS + c0 + k) * 2;
    s_init[idx] = s1;
    s_init[idx + 1] = s2;
    float d1 = dbase[(c0 + k) * 2], d2 = dbase[(c0 + k) * 2 + 1];
    float n1 = fmaf(M00, s1, fmaf(M01, s2, d1));
    float n2 = fmaf(M10, s1, fmaf(M11, s2, d2));
    s1 = n1; s2 = n2;
  }
}

// ============ Kernel 3: replay chunks with true state, emit y ============
__global__ void k3_emit(const float* __restrict__ x,
                        const float* __restrict__ ff,
                        const float* __restrict__ fq,
                        const float* __restrict__ s_init,
                        float* __restrict__ y) {
  extern __shared__ float smem[]; // [in0 | in1 | out0 | out1]
  const int lane = threadIdx.x;
  const int b = blockIdx.x / WPR;
  const int w = blockIdx.x % WPR;
  const int c = w * ROWS + lane;

  float b0, b1, b2, a1, a2;
  lp_coeffs(ff[0], fq[0], b0, b1, b2, a1, a2);
  const float na1 = -a1, na2 = -a2;

  const uint64_t rowBase = (uint64_t)b * TLEN;
  const uint64_t chunkStart = rowBase + (uint64_t)c * LCH;
  float x1 = 0.f, x2 = 0.f;
  if (c > 0) { x1 = x[chunkStart - 1]; x2 = x[chunkStart - 2]; }
  const int sidx = (b * CHUNKS + c) * 2;
  float y1 = s_init[sidx], y2 = s_init[sidx + 1];

#if HAVE_TDM
  const uint64_t base = rowBase + (uint64_t)(w * ROWS) * LCH;
  const uint64_t xbase = (uint64_t)(uintptr_t)x + 4ull * base;
  const uint64_t ybase = (uint64_t)(uintptr_t)y + 4ull * base;
  tdm_load(xbase, 0u);
  for (int t = 0; t < NTILES; ++t) {
    if (t + 1 < NTILES) {
      asm volatile("s_wait_dscnt 0" ::: "memory");
      tdm_load(xbase + 4ull * (uint64_t)(t + 1) * TILE,
               ((t + 1) & 1) ? (uint32_t)TILE_BYTES : 0u);
      __builtin_amdgcn_s_wait_tensorcnt((short)1); // in-order: Ld(t) & St(t-1) retired
    } else {
      __builtin_amdgcn_s_wait_tensorcnt((short)0);
    }
    asm volatile("" ::: "memory");
    const float4* bv = (const float4*)(smem + (t & 1) * TILE_FLOATS + lane * TILE);
    float4* ov = (float4*)(smem + (2 + (t & 1)) * TILE_FLOATS + lane * TILE);
    for (int j = 0; j < TILE / 4; ++j) {
      float4 v = bv[j];
      float4 o;
      BIQ_STEP(v.x); o.x = y1;
      BIQ_STEP(v.y); o.y = y1;
      BIQ_STEP(v.z); o.z = y1;
      BIQ_STEP(v.w); o.w = y1;
      ov[j] = o;
    }
    asm volatile("s_wait_dscnt 0" ::: "memory"); // LDS writes visible to TDM
    tdm_store(ybase + 4ull * (uint64_t)t * TILE, (uint32_t)((2 + (t & 1)) * TILE_BYTES));
  }
  __builtin_amdgcn_s_wait_tensorcnt((short)0);
#else
  const float* gx = x + chunkStart;
  float* gy = y + chunkStart;
  for (int n = 0; n < LCH; ++n) {
    float xt = gx[n];
    BIQ_STEP(xt);
    gy[n] = y1;
  }
#endif
}

// ============ Fallback if workspace is too small: fully sequential ============
__global__ void k_naive(const float* __restrict__ x,
                        const float* __restrict__ ff,
                        const float* __restrict__ fq,
                        float* __restrict__ y) {
  const int b = threadIdx.x;
  if (b >= BATCH) return;
  float b0, b1, b2, a1, a2;
  lp_coeffs(ff[0], fq[0], b0, b1, b2, a1, a2);
  const float na1 = -a1, na2 = -a2;
  float x1 = 0.f, x2 = 0.f, y1 = 0.f, y2 = 0.f;
  const float* gx = x + (size_t)b * TLEN;
  float* gy = y + (size_t)b * TLEN;
  for (int n = 0; n < TLEN; ++n) {
    float xt = gx[n];
    BIQ_STEP(xt);
    gy[n] = y1;
  }
}

// ---------------- host entry ----------------
extern "C" void kernel_launch(void* const* d_in, const int* in_sizes, int n_in,
                              void* d_out, int out_size, void* d_ws, size_t ws_size,
                              hipStream_t stream) {
  (void)in_sizes; (void)n_in; (void)out_size;
  const float* x  = (const float*)d_in[0];
  const float* ff = (const float*)d_in[2];
  const float* fq = (const float*)d_in[3];
  float* y = (float*)d_out;

  const size_t ws_need = (size_t)4 * BATCH * CHUNKS * sizeof(float);
  if (ws_size < ws_need) {
    k_naive<<<dim3(1), dim3(BATCH), 0, stream>>>(x, ff, fq, y);
    return;
  }

  float* d_state = (float*)d_ws;                       // 2*B*C floats
  float* s_init  = d_state + (size_t)2 * BATCH * CHUNKS; // 2*B*C floats

  dim3 grid(BATCH * WPR), block(ROWS);
  k1_chunk_state<<<grid, block, 2 * TILE_BYTES, stream>>>(x, ff, fq, d_state);
  k2_prefix<<<dim3(BATCH), dim3(32), 0, stream>>>(ff, fq, d_state, s_init);
  k3_emit<<<grid, block, 4 * TILE_BYTES, stream>>>(x, ff, fq, s_init, y);
}